// FanBackproj2D_29850022707686
// MI455X (gfx1250) — compile-verified
//
#include <hip/hip_runtime.h>
#include <hip/hip_bf16.h>
#include <stdint.h>

// Fan-beam backprojection, CDNA5 (gfx1250, wave32).
//   sino: [B=4, A=360, D=1024] f32  ->  out: [B=4, H=512, W=512] f32
//
// 16x16 pixel tile per 256-thread block (8 wave32s). Sinogram rows are staged
// into LDS with global_load_async_to_lds_b128 (ASYNCcnt-tracked DMA), double
// buffered with TWO angles per stage (64 KB of the 320 KB/WGP LDS) so each
// barrier pair covers two angles and 16 B128 DMAs/thread are in flight.
// Geometry is computed once per pixel per angle and reused for all 4 batches.

#define A_NUM   360
#define D_NUM   1024
#define B_NUM   4
#define HW      512
#define TILE    16

#define SID_F   750.0f
#define SDD_F   1200.0f
#define DBETA   0.017453292519943295f   // 2*pi/360

#define ROW_FLOATS     (B_NUM * D_NUM)           // 4096 floats per angle (4 batches)
#define ROWSET_BYTES   (ROW_FLOATS * 4)          // 16384 B per angle
#define STAGE_FLOATS   (2 * ROW_FLOATS)          // two angles per stage
#define STAGE_BYTES    (2 * ROWSET_BYTES)        // 32768 B per stage
#define BATCH_STRIDE_B (A_NUM * D_NUM * 4)       // 1474560 B between batches

// Issue the 4 batch rows of one angle into LDS slot (stage, slot01).
__device__ __forceinline__ void async_row_loads(const float* sino_base,
                                                uint32_t lds_byte_base,
                                                int stage, int slot01, int a, int tid) {
    uint32_t lds0  = lds_byte_base +
                     (uint32_t)(stage * STAGE_BYTES + slot01 * ROWSET_BYTES + tid * 16);
    uint32_t goff0 = (uint32_t)(a * (D_NUM * 4) + tid * 16);
#pragma unroll
    for (int b = 0; b < B_NUM; ++b) {
        uint32_t lds  = lds0  + (uint32_t)(b * (D_NUM * 4));
        uint32_t goff = goff0 + (uint32_t)(b * BATCH_STRIDE_B);
        asm volatile("global_load_async_to_lds_b128 %0, %1, %2"
                     :
                     : "v"(lds), "v"(goff), "s"(sino_base)
                     : "memory");
    }
}

__global__ __launch_bounds__(256)
void fan_backproj_kernel(const float* __restrict__ sino,
                         float* __restrict__ out) {
    __shared__ __align__(16) float smem[2 * STAGE_FLOATS]; // 64 KB double buffer
    __shared__ float2 cs_tab[A_NUM];                       // cos/sin per angle

    const int tid = (int)threadIdx.x;            // 0..255
    const int px  = tid & (TILE - 1);
    const int py  = tid >> 4;
    const int x   = (int)blockIdx.x * TILE + px;
    const int y   = (int)blockIdx.y * TILE + py;

    // angle table (in LDS, computed once per block)
    for (int i = tid; i < A_NUM; i += 256) {
        float beta = (float)i * DBETA;
        float s, c;
        __sincosf(beta, &s, &c);
        cs_tab[i] = make_float2(c, s);
    }

    const float X = ((float)x - 255.5f);
    const float Y = ((float)y - 255.5f);

    float acc0 = 0.f, acc1 = 0.f, acc2 = 0.f, acc3 = 0.f;

    const uint32_t smem_base = (uint32_t)(uintptr_t)(&smem[0]); // LDS byte addr

    // prologue: prefetch stages for angles {0,1} and {2,3}
    async_row_loads(sino, smem_base, 0, 0, 0, tid);
    async_row_loads(sino, smem_base, 0, 1, 1, tid);
    async_row_loads(sino, smem_base, 1, 0, 2, tid);
    async_row_loads(sino, smem_base, 1, 1, 3, tid);

    for (int a = 0; a < A_NUM; a += 2) {
        const int stage = (a >> 1) & 1;

        if (a + 2 < A_NUM) {
            // in-order completion: <=8 outstanding => this stage's 8 DMAs done
            asm volatile("s_wait_asynccnt 8" ::: "memory");
        } else {
            asm volatile("s_wait_asynccnt 0" ::: "memory");
        }
        __syncthreads();   // all waves' DMA done (and angle table visible)

#pragma unroll
        for (int k = 0; k < 2; ++k) {
            const float2 cs = cs_tab[a + k];
            const float c  = cs.x;
            const float sn = cs.y;

            float t = SID_F + X * c + Y * sn;          // > 0 for this geometry
            float r = __builtin_amdgcn_rcpf(t);
            r = r * (2.0f - t * r);                    // one Newton step
            float u    = SDD_F * (Y * c - X * sn) * r; // flat-detector coord
            float det  = u * 0.5f + 511.5f;            // u/DS + (D-1)/2
            float i0f  = floorf(det);
            float frac = det - i0f;
            int   i0   = (int)i0f;
            i0 = (i0 < 0) ? 0 : ((i0 > D_NUM - 2) ? (D_NUM - 2) : i0);
            bool valid = (det >= 0.0f) & (det <= (float)(D_NUM - 1)) & (t > 0.0f);
            float sr = SID_F * r;
            float w  = valid ? (sr * sr) : 0.0f;

            const float* buf = &smem[stage * STAGE_FLOATS + k * ROW_FLOATS];
            float v0 = buf[0 * D_NUM + i0], v1 = buf[0 * D_NUM + i0 + 1];
            acc0 += w * (v0 + frac * (v1 - v0));
            v0 = buf[1 * D_NUM + i0]; v1 = buf[1 * D_NUM + i0 + 1];
            acc1 += w * (v0 + frac * (v1 - v0));
            v0 = buf[2 * D_NUM + i0]; v1 = buf[2 * D_NUM + i0 + 1];
            acc2 += w * (v0 + frac * (v1 - v0));
            v0 = buf[3 * D_NUM + i0]; v1 = buf[3 * D_NUM + i0 + 1];
            acc3 += w * (v0 + frac * (v1 - v0));
        }

        __syncthreads();   // everyone done reading this stage buffer

        if (a + 4 < A_NUM) {
            async_row_loads(sino, smem_base, stage, 0, a + 4, tid);
            async_row_loads(sino, smem_base, stage, 1, a + 5, tid);
        }
    }

    const int pix = y * HW + x;
    out[0 * (HW * HW) + pix] = acc0 * DBETA;
    out[1 * (HW * HW) + pix] = acc1 * DBETA;
    out[2 * (HW * HW) + pix] = acc2 * DBETA;
    out[3 * (HW * HW) + pix] = acc3 * DBETA;
}

extern "C" void kernel_launch(void* const* d_in, const int* in_sizes, int n_in,
                              void* d_out, int out_size, void* d_ws, size_t ws_size,
                              hipStream_t stream) {
    (void)in_sizes; (void)n_in; (void)d_ws; (void)ws_size; (void)out_size;
    const float* sino = (const float*)d_in[0];   // [4,360,1024,1] f32
    float*       out  = (float*)d_out;           // [4,512,512,1] f32

    dim3 grid(HW / TILE, HW / TILE, 1);          // 32 x 32 blocks
    dim3 block(256, 1, 1);                       // 8 wave32s
    fan_backproj_kernel<<<grid, block, 0, stream>>>(sino, out);
}